// EAEF_87101936763064
// MI455X (gfx1250) — compile-verified
//
#include <hip/hip_runtime.h>
#include <math.h>

#define PB 16      // batch
#define PN 4096    // input points
#define PM 512     // fps samples
#define PK 16      // knn k

typedef float v2f __attribute__((ext_vector_type(2)));
typedef float v8f __attribute__((ext_vector_type(8)));

// D = A(16x4) * B(4x16) + C, fp32 WMMA (CDNA5, wave32)
__device__ __forceinline__ v8f wmma_f32x4(v2f a, v2f b, v8f c) {
  return __builtin_amdgcn_wmma_f32_16x16x4_f32(false, a, false, b, (short)0, c, false, false);
}

// ---------------------------------------------------------------------------
// Weight prep: Wqk = Wq - Wk and Wv, zero-padded from Cw to Kpad columns.
// Removes all guards / redundant subtractions from the hot WMMA loops.
// ---------------------------------------------------------------------------
__global__ __launch_bounds__(256)
void prep_weights_kernel(const float* __restrict__ Wq, const float* __restrict__ Wk,
                         const float* __restrict__ Wv, float* __restrict__ outQK,
                         float* __restrict__ outV, int D, int Cw, int Kpad) {
  int i = blockIdx.x * 256 + threadIdx.x;
  if (i >= D * Kpad) return;
  int d = i / Kpad, k = i % Kpad;
  float q = 0.f, v = 0.f;
  if (k < Cw) {
    q = Wq[(size_t)d * Cw + k] - Wk[(size_t)d * Cw + k];
    v = Wv[(size_t)d * Cw + k];
  }
  outQK[i] = q;
  outV[i] = v;
}

// ---------------------------------------------------------------------------
// Farthest point sampling: x[B,N,3] -> pos[B,3,M]  (one block per batch)
// ---------------------------------------------------------------------------
__global__ __launch_bounds__(256)
void fps_kernel(const float* __restrict__ x, float* __restrict__ pos) {
  const int b = blockIdx.x;
  const int tid = threadIdx.x;
  const int PPT = PN / 256; // 16 points per thread
  __shared__ float sLast[3];
  __shared__ int sSel[PM];
  __shared__ float sVal[256];
  __shared__ int sArg[256];

  float px[PPT], py[PPT], pz[PPT], dmin[PPT];
#pragma unroll
  for (int i = 0; i < PPT; ++i) {
    int p = tid + i * 256;
    const float* xp = x + ((size_t)b * PN + p) * 3;
    px[i] = xp[0]; py[i] = xp[1]; pz[i] = xp[2];
    dmin[i] = INFINITY;
  }
  if (tid == 0) {
    const float* xp = x + (size_t)b * PN * 3;
    sLast[0] = xp[0]; sLast[1] = xp[1]; sLast[2] = xp[2];
    sSel[0] = 0;
  }
  __syncthreads();
  for (int it = 1; it < PM; ++it) {
    float bx = sLast[0], by = sLast[1], bz = sLast[2];
    float best = -INFINITY; int barg = PN;
#pragma unroll
    for (int i = 0; i < PPT; ++i) {
      float dx = px[i] - bx, dy = py[i] - by, dz = pz[i] - bz;
      float d = dx * dx + dy * dy + dz * dz;
      dmin[i] = fminf(dmin[i], d);
      int p = tid + i * 256;
      if (dmin[i] > best || (dmin[i] == best && p < barg)) { best = dmin[i]; barg = p; }
    }
    sVal[tid] = best; sArg[tid] = barg;
    __syncthreads();
    for (int s = 128; s > 0; s >>= 1) {
      if (tid < s) {
        float v = sVal[tid + s]; int a = sArg[tid + s];
        if (v > sVal[tid] || (v == sVal[tid] && a < sArg[tid])) { sVal[tid] = v; sArg[tid] = a; }
      }
      __syncthreads();
    }
    if (tid == 0) {
      int nxt = sArg[0];
      sSel[it] = nxt;
      const float* xp = x + ((size_t)b * PN + nxt) * 3;
      sLast[0] = xp[0]; sLast[1] = xp[1]; sLast[2] = xp[2];
    }
    __syncthreads();
  }
  for (int i = tid; i < PM; i += 256) {
    int p = sSel[i];
    const float* xp = x + ((size_t)b * PN + p) * 3;
    pos[((size_t)b * 3 + 0) * PM + i] = xp[0];
    pos[((size_t)b * 3 + 1) * PM + i] = xp[1];
    pos[((size_t)b * 3 + 2) * PM + i] = xp[2];
  }
}

// ---------------------------------------------------------------------------
// kNN: f[B,C,M] -> idx[B,M,K] (K smallest squared distances, self included)
// one block per (b, query point); order irrelevant (attention is perm-invariant)
// ---------------------------------------------------------------------------
__global__ __launch_bounds__(256)
void knn_kernel(const float* __restrict__ f, int C, int* __restrict__ idxout) {
  const int bn = blockIdx.x;
  const int b = bn / PM, n = bn % PM;
  const int tid = threadIdx.x;
  __shared__ float sD[PM];
  __shared__ float sVal[256];
  __shared__ int sArg[256];
  const float* fb = f + (size_t)b * C * PM;
  for (int m = tid; m < PM; m += 256) {
    float acc = 0.f;
    for (int c = 0; c < C; ++c) {
      float d = fb[(size_t)c * PM + n] - fb[(size_t)c * PM + m];
      acc += d * d;
    }
    sD[m] = acc;
  }
  __syncthreads();
  for (int r = 0; r < PK; ++r) {
    float best = INFINITY; int barg = PM;
    for (int m = tid; m < PM; m += 256) {
      float v = sD[m];
      if (v < best || (v == best && m < barg)) { best = v; barg = m; }
    }
    sVal[tid] = best; sArg[tid] = barg;
    __syncthreads();
    for (int s = 128; s > 0; s >>= 1) {
      if (tid < s) {
        float v = sVal[tid + s]; int a = sArg[tid + s];
        if (v < sVal[tid] || (v == sVal[tid] && a < sArg[tid])) { sVal[tid] = v; sArg[tid] = a; }
      }
      __syncthreads();
    }
    if (tid == 0) {
      idxout[(size_t)bn * PK + r] = sArg[0];
      sD[sArg[0]] = INFINITY;
    }
    __syncthreads();
  }
}

// ---------------------------------------------------------------------------
// Fully fused point-transformer layer, ONE query point per block.
//   h  = Wqk @ gf + pe   (WMMA f32 16x16x4; Wqk = Wq-Wk prepadded)
//   v' = Wv  @ gf + pe
//   t  = relu(A1 @ h)
//   l  = A2 @ t
//   out[d] = sum_j softmax_j(l[d,:]) * v'[d,j]
// B-operand tiles (gf, h, t) are column-major in LDS with +2 row pad:
//   one ds_load_b64 per lane per WMMA, 16 lanes land on distinct banks.
// A-operand loads are contiguous v2f -> global_load_b64.
// LDS aliasing: [rel | hT | v' | {gfT,pe -> tT} | l | idx]  (<= ~58 KB)
// ---------------------------------------------------------------------------
__global__ __launch_bounds__(256)
void xform_kernel(const float* __restrict__ f, const float* __restrict__ pos,
                  const int* __restrict__ idx,
                  const float* __restrict__ WqkP, const float* __restrict__ WvP,
                  const float* __restrict__ P1, const float* __restrict__ P2,
                  const float* __restrict__ A1, const float* __restrict__ A2,
                  float* __restrict__ out, int Cin, int Kpad, int D, int H) {
  const int bn = blockIdx.x;
  const int b = bn / PM, n = bn % PM;
  const int tid = threadIdx.x;
  const int lane = tid & 31, wave = tid >> 5;
  const int SG = Kpad + 2;  // col strides (+2 pad: bank step 2 across lanes)
  const int SD = D + 2;
  const int SHS = H + 2;
  extern __shared__ float smem[];
  const int scrF = (16 * SG + D * 16 > 16 * SHS) ? (16 * SG + D * 16) : (16 * SHS);
  float* sRel = smem;                  // 3*16
  float* sHT  = smem + 48;             // 16*SD  col-major h (phase A: t1)
  float* sV   = sHT + 16 * SD;         // D*16   row-major v'
  float* sScr = sV + D * 16;           // scrF
  float* sGfT = sScr;                  // 16*SG  col-major gf (phase A)
  float* sPe  = sScr + 16 * SG;        // D*16   row-major pe (phase A)
  float* sTT  = sScr;                  // 16*SHS col-major t  (phase B)
  float* sL   = sScr + scrF;           // D*16   row-major logits
  int*   sNb  = (int*)(sL + D * 16);   // 16

  if (tid == 0) {                      // warm GL2 for the big weight matrices
    __builtin_prefetch(A1, 0, 3);
    __builtin_prefetch(A2, 0, 3);
  }
  if (tid < PK) sNb[tid] = idx[(size_t)bn * PK + tid];
  __syncthreads();

  // relative positions + edge features (cols j, rows k; pad rows zeroed)
  if (tid < 48) {
    int c = tid >> 4, j = tid & 15;
    sRel[c * 16 + j] = pos[((size_t)b * 3 + c) * PM + sNb[j]] - pos[((size_t)b * 3 + c) * PM + n];
  }
  for (int e = tid; e < 16 * Kpad; e += 256) {
    int j = e / Kpad, row = e % Kpad;
    float v = 0.f;
    if (row < Cin) {
      v = f[((size_t)b * Cin + row) * PM + sNb[j]] - f[((size_t)b * Cin + row) * PM + n];
    } else if (row < 2 * Cin) {
      int c = row - Cin;
      v = f[((size_t)b * Cin + c) * PM + n];
    }
    sGfT[j * SG + row] = v;
  }
  __syncthreads();

  // t1 = relu(P1 @ rel) -> sHT (col-major temp, overwritten by h later)
  for (int e = tid; e < D * 16; e += 256) {
    int d = e % D, j = e / D;
    float a = P1[d * 3 + 0] * sRel[j] + P1[d * 3 + 1] * sRel[16 + j] + P1[d * 3 + 2] * sRel[32 + j];
    sHT[j * SD + d] = fmaxf(a, 0.f);
  }
  __syncthreads();
  // pe = P2 @ t1 (row-major out; t1 contiguous in q)
  for (int e = tid; e < D * 16; e += 256) {
    int d = e >> 4, j = e & 15;
    float acc = 0.f;
    for (int q = 0; q < D; ++q) acc += P2[(size_t)d * D + q] * sHT[j * SD + q];
    sPe[d * 16 + j] = acc;
  }
  __syncthreads();

  // WMMA lane mapping (ISA 7.12.2):
  //  A 16x4: lanes 0-15 row=lane {K0,K1}; lanes 16-31 row=lane-16 {K2,K3}
  //  B 4x16 / C 16x16: col = lane&15; K/M rows split low/high half-wave
  const int r = lane & 15;
  const int khalf = (lane >> 4) << 1; // 0 or 2
  const int rhalf = (lane >> 4) << 3; // 0 or 8

  // h = Wqk @ gf + pe ;  v' = Wv @ gf + pe
  for (int rc = wave; rc < D / 16; rc += 8) {
    int rowBase = rc * 16;
    int row = rowBase + r;
    v8f acc, accv;
#pragma unroll
    for (int u = 0; u < 8; ++u) {
      float pe = sPe[(rowBase + u + rhalf) * 16 + r];
      acc[u] = pe; accv[u] = pe;
    }
#pragma unroll 4
    for (int kk = 0; kk < Kpad; kk += 4) {
      int ka = kk + khalf;
      v2f bfrag = *(const v2f*)(sGfT + r * SG + ka);
      v2f aq    = *(const v2f*)(WqkP + (size_t)row * Kpad + ka);
      v2f av    = *(const v2f*)(WvP  + (size_t)row * Kpad + ka);
      acc  = wmma_f32x4(aq, bfrag, acc);
      accv = wmma_f32x4(av, bfrag, accv);
    }
#pragma unroll
    for (int u = 0; u < 8; ++u) {
      sHT[r * SD + rowBase + u + rhalf] = acc[u];           // col-major (B of next GEMM)
      sV[(rowBase + u + rhalf) * 16 + r] = accv[u];         // row-major (softmax)
    }
  }
  __syncthreads();

  // t = relu(A1 @ h) -> sTT (aliases gfT/pe, both dead)
  for (int rc = wave; rc < H / 16; rc += 8) {
    int rowBase = rc * 16;
    int row = rowBase + r;
    v8f acc;
#pragma unroll
    for (int u = 0; u < 8; ++u) acc[u] = 0.f;
#pragma unroll 4
    for (int kk = 0; kk < D; kk += 4) {
      int ka = kk + khalf;
      v2f bfrag = *(const v2f*)(sHT + r * SD + ka);
      v2f afrag = *(const v2f*)(A1 + (size_t)row * D + ka);
      acc = wmma_f32x4(afrag, bfrag, acc);
    }
#pragma unroll
    for (int u = 0; u < 8; ++u)
      sTT[r * SHS + rowBase + u + rhalf] = fmaxf(acc[u], 0.f);
  }
  __syncthreads();

  // logits = A2 @ t -> sL
  for (int rc = wave; rc < D / 16; rc += 8) {
    int rowBase = rc * 16;
    int row = rowBase + r;
    v8f acc;
#pragma unroll
    for (int u = 0; u < 8; ++u) acc[u] = 0.f;
#pragma unroll 4
    for (int kk = 0; kk < H; kk += 4) {
      int ka = kk + khalf;
      v2f bfrag = *(const v2f*)(sTT + r * SHS + ka);
      v2f afrag = *(const v2f*)(A2 + (size_t)row * H + ka);
      acc = wmma_f32x4(afrag, bfrag, acc);
    }
#pragma unroll
    for (int u = 0; u < 8; ++u)
      sL[(rowBase + u + rhalf) * 16 + r] = acc[u];
  }
  __syncthreads();

  // softmax over neighbors, reduce with v'
  for (int d = tid; d < D; d += 256) {
    float mx = -INFINITY;
#pragma unroll
    for (int j = 0; j < PK; ++j) mx = fmaxf(mx, sL[d * 16 + j]);
    float s = 0.f, o = 0.f;
#pragma unroll
    for (int j = 0; j < PK; ++j) {
      float p = __expf(sL[d * 16 + j] - mx);
      s += p;
      o += p * sV[d * 16 + j];
    }
    out[((size_t)b * D + d) * PM + n] = o / s;
  }
}

// ---------------------------------------------------------------------------
// feat = concat(x1,x2,x3) [B,256,M]; out = [max_M(feat) | mean_M(feat)]
// ---------------------------------------------------------------------------
__global__ __launch_bounds__(256)
void pool_kernel(const float* __restrict__ x1, const float* __restrict__ x2,
                 const float* __restrict__ x3, float* __restrict__ out) {
  int b = blockIdx.x, ch = threadIdx.x;
  const float* src;
  if (ch < 64)       src = x1 + ((size_t)b * 64 + ch) * PM;
  else if (ch < 128) src = x2 + ((size_t)b * 64 + (ch - 64)) * PM;
  else               src = x3 + ((size_t)b * 128 + (ch - 128)) * PM;
  float mx = -INFINITY, sm = 0.f;
  for (int m = 0; m < PM; ++m) { float v = src[m]; mx = fmaxf(mx, v); sm += v; }
  out[(size_t)b * 512 + ch] = mx;
  out[(size_t)b * 512 + 256 + ch] = sm * (1.f / PM);
}

// ---------------------------------------------------------------------------
extern "C" void kernel_launch(void* const* d_in, const int* in_sizes, int n_in,
                              void* d_out, int out_size, void* d_ws, size_t ws_size,
                              hipStream_t stream) {
  (void)in_sizes; (void)n_in; (void)out_size; (void)ws_size;
  const float* x = (const float*)d_in[0];
  const float* t1[7]; const float* t2[7]; const float* t3[7];
  for (int i = 0; i < 7; ++i) {
    t1[i] = (const float*)d_in[1 + i];
    t2[i] = (const float*)d_in[8 + i];
    t3[i] = (const float*)d_in[15 + i];
  }
  float* ws = (float*)d_ws;
  size_t off = 0;
  float* pos = ws + off; off += (size_t)PB * 3 * PM;
  float* x1  = ws + off; off += (size_t)PB * 64 * PM;
  float* x2  = ws + off; off += (size_t)PB * 64 * PM;
  float* x3  = ws + off; off += (size_t)PB * 128 * PM;
  int* idxb  = (int*)(ws + off); off += (size_t)PB * PM * PK;
  float* wqk1 = ws + off; off += 64 * 8;
  float* wv1  = ws + off; off += 64 * 8;
  float* wqk2 = ws + off; off += 64 * 128;
  float* wv2  = ws + off; off += 64 * 128;
  float* wqk3 = ws + off; off += 128 * 128;
  float* wv3  = ws + off; off += 128 * 128;

  auto lds = [](int Kpad, int D, int H) -> size_t {
    int scr = (16 * (Kpad + 2) + D * 16 > 16 * (H + 2)) ? (16 * (Kpad + 2) + D * 16)
                                                        : (16 * (H + 2));
    return ((size_t)(48 + 16 * (D + 2) + D * 16 + scr + D * 16)) * sizeof(float) +
           PK * sizeof(int);
  };

  // weight prep (tiny)
  prep_weights_kernel<<<(64 * 8 + 255) / 256, 256, 0, stream>>>(
      t1[0], t1[1], t1[2], wqk1, wv1, 64, 6, 8);
  prep_weights_kernel<<<(64 * 128 + 255) / 256, 256, 0, stream>>>(
      t2[0], t2[1], t2[2], wqk2, wv2, 64, 128, 128);
  prep_weights_kernel<<<(128 * 128 + 255) / 256, 256, 0, stream>>>(
      t3[0], t3[1], t3[2], wqk3, wv3, 128, 128, 128);

  fps_kernel<<<PB, 256, 0, stream>>>(x, pos);

  // layer 1: Cin=3 (2C=6 padded to 8), D=64, H=256
  knn_kernel<<<PB * PM, 256, 0, stream>>>(pos, 3, idxb);
  xform_kernel<<<PB * PM, 256, lds(8, 64, 256), stream>>>(
      pos, pos, idxb, wqk1, wv1, t1[3], t1[4], t1[5], t1[6],
      x1, 3, 8, 64, 256);

  // layer 2: Cin=64 -> 2C=128, D=64, H=256
  knn_kernel<<<PB * PM, 256, 0, stream>>>(x1, 64, idxb);
  xform_kernel<<<PB * PM, 256, lds(128, 64, 256), stream>>>(
      x1, pos, idxb, wqk2, wv2, t2[3], t2[4], t2[5], t2[6],
      x2, 64, 128, 64, 256);

  // layer 3: Cin=64 -> 2C=128, D=128, H=512
  knn_kernel<<<PB * PM, 256, 0, stream>>>(x2, 64, idxb);
  xform_kernel<<<PB * PM, 256, lds(128, 128, 512), stream>>>(
      x2, pos, idxb, wqk3, wv3, t3[3], t3[4], t3[5], t3[6],
      x3, 64, 128, 128, 512);

  pool_kernel<<<PB, 256, 0, stream>>>(x1, x2, x3, (float*)d_out);
}